// Net1_36335423324471
// MI455X (gfx1250) — compile-verified
//
#include <hip/hip_runtime.h>
#include <hip/hip_bf16.h>
#include <math.h>

#define NNODES 100000
#define NEDGES 3200000
#define FIN    128
#define DIM    32
#define FCIN   (FIN + 3*DIM)   // 224
#define NCLS   10

typedef float v2f __attribute__((ext_vector_type(2)));
typedef float v8f __attribute__((ext_vector_type(8)));

// ---------------- degree: deg[col[e]] += w[e] ----------------
__global__ void k_deg(const int* __restrict__ col, const float* __restrict__ w,
                      float* __restrict__ deg) {
    int e = blockIdx.x * blockDim.x + threadIdx.x;
    if (e < NEDGES) unsafeAtomicAdd(&deg[col[e]], w[e]);
}

// ---------------- dis = deg>0 ? rsqrt(deg) : 0 (in place) ----------------
__global__ void k_dis(float* __restrict__ deg) {
    int i = blockIdx.x * blockDim.x + threadIdx.x;
    if (i < NNODES) {
        float d = deg[i];
        deg[i] = (d > 0.0f) ? rsqrtf(d) : 0.0f;
    }
}

// ---------------- norm[e] = dis[row]*w*dis[col] (layer-invariant) ----------------
__global__ void k_norm(const int* __restrict__ row, const int* __restrict__ col,
                       const float* __restrict__ w, const float* __restrict__ dis,
                       float* __restrict__ norm) {
    int e = blockIdx.x * blockDim.x + threadIdx.x;
    if (e < NEDGES) norm[e] = dis[row[e]] * w[e] * dis[col[e]];
}

// ---------------- copy x into concat[:, 0:128] ----------------
__global__ void k_copyx(const float* __restrict__ x, float* __restrict__ cc) {
    int id = blockIdx.x * blockDim.x + threadIdx.x;   // N*FIN
    if (id < NNODES * FIN) {
        int n = id >> 7, f = id & 127;
        cc[(size_t)n * FCIN + f] = x[id];
    }
}

// ---------------- H[N,32] = A[N,K](lda) @ W[K,32] via V_WMMA_F32_16X16X4_F32 ----
// One wave computes a 16x32 output tile (two 16x16 accumulators).
__global__ void k_gemm_wmma(const float* __restrict__ A, int lda, int K,
                            const float* __restrict__ W, float* __restrict__ H) {
    const int lane = threadIdx.x & 31;
    const int wid  = threadIdx.x >> 5;
    const int tile = blockIdx.x * (blockDim.x >> 5) + wid;
    if (tile * 16 >= NNODES) return;                 // whole-wave exit, EXEC stays full

    const int m  = lane & 15;                        // A row within tile / B,C col
    const int kh = (lane >> 4) << 1;                 // K sub-offset: 0 or 2
    const float* __restrict__ arow = A + (size_t)(tile * 16 + m) * lda;

    v8f acc0 = {};   // cols 0..15
    v8f acc1 = {};   // cols 16..31
    for (int k = 0; k < K; k += 4) {
        const int kk = k + kh;
        v2f a;  a.x  = arow[kk];                 a.y  = arow[kk + 1];
        v2f b0; b0.x = W[kk * DIM + m];          b0.y = W[(kk + 1) * DIM + m];
        v2f b1; b1.x = W[kk * DIM + 16 + m];     b1.y = W[(kk + 1) * DIM + 16 + m];
        acc0 = __builtin_amdgcn_wmma_f32_16x16x4_f32(false, a, false, b0,
                                                     (short)0, acc0, false, false);
        acc1 = __builtin_amdgcn_wmma_f32_16x16x4_f32(false, a, false, b1,
                                                     (short)0, acc1, false, false);
    }
    // D layout: VGPR r -> row r (lanes 0-15) / row 8+r (lanes 16-31); col = lane&15
    const int mbase = tile * 16 + ((lane >> 4) << 3);
#pragma unroll
    for (int r = 0; r < 8; ++r) {
        H[(size_t)(mbase + r) * DIM + m]      = acc0[r];
        H[(size_t)(mbase + r) * DIM + 16 + m] = acc1[r];
    }
}

// ---------------- aggregation: wave per edge, lane = feature ----------------
__global__ void k_agg(const int* __restrict__ row, const int* __restrict__ col,
                      const float* __restrict__ norm, const float* __restrict__ H,
                      float* __restrict__ O) {
    const int lane = threadIdx.x & 31;
    const long long e = (long long)blockIdx.x * (blockDim.x >> 5) + (threadIdx.x >> 5);
    if (e >= NEDGES) return;
    const int r = row[e], c = col[e];
    const float v = H[(size_t)r * DIM + lane] * norm[e];
    unsafeAtomicAdd(&O[(size_t)c * DIM + lane], v);
}

// ---------------- bias + relu, write into concat slice ----------------
__global__ void k_bias_relu(const float* __restrict__ O, const float* __restrict__ b,
                            float* __restrict__ cslice) {
    int id = blockIdx.x * blockDim.x + threadIdx.x;   // N*DIM
    if (id < NNODES * DIM) {
        int n = id >> 5, f = id & 31;
        float v = O[id] + b[f];
        cslice[(size_t)n * FCIN + f] = fmaxf(v, 0.0f);
    }
}

// ---------------- FC head + log_softmax: wave per node ----------------
__global__ void k_fc(const float* __restrict__ cc, const float* __restrict__ fcW,
                     const float* __restrict__ fcb, float* __restrict__ out) {
    const int lane = threadIdx.x & 31;
    const int n = blockIdx.x * (blockDim.x >> 5) + (threadIdx.x >> 5);
    if (n >= NNODES) return;

    float acc[NCLS];
#pragma unroll
    for (int c = 0; c < NCLS; ++c) acc[c] = 0.0f;

    const float* __restrict__ rowp = cc + (size_t)n * FCIN;
    for (int k = lane; k < FCIN; k += 32) {
        const float xv = rowp[k];
#pragma unroll
        for (int c = 0; c < NCLS; ++c) acc[c] += xv * fcW[k * NCLS + c];
    }
#pragma unroll
    for (int c = 0; c < NCLS; ++c) {
#pragma unroll
        for (int off = 16; off > 0; off >>= 1)
            acc[c] += __shfl_xor(acc[c], off, 32);
    }
    float mx = -INFINITY;
#pragma unroll
    for (int c = 0; c < NCLS; ++c) { acc[c] += fcb[c]; mx = fmaxf(mx, acc[c]); }
    float s = 0.0f;
#pragma unroll
    for (int c = 0; c < NCLS; ++c) s += __expf(acc[c] - mx);
    const float lse = mx + __logf(s);
    if (lane < NCLS) out[(size_t)n * NCLS + lane] = acc[lane] - lse;
}

extern "C" void kernel_launch(void* const* d_in, const int* in_sizes, int n_in,
                              void* d_out, int out_size, void* d_ws, size_t ws_size,
                              hipStream_t stream) {
    const float* x    = (const float*)d_in[0];
    const int*   eidx = (const int*)d_in[1];          // [2, E] flattened
    const float* ew   = (const float*)d_in[2];
    const float* W0   = (const float*)d_in[3];
    const float* b0   = (const float*)d_in[4];
    const float* W1   = (const float*)d_in[5];
    const float* b1   = (const float*)d_in[6];
    const float* W2   = (const float*)d_in[7];
    const float* b2   = (const float*)d_in[8];
    const float* fcW  = (const float*)d_in[9];
    const float* fcb  = (const float*)d_in[10];
    float* out = (float*)d_out;

    const int* row = eidx;            // edge_index[0]
    const int* col = eidx + NEDGES;   // edge_index[1]

    // workspace carve-up
    char* ws = (char*)d_ws;
    float* norm   = (float*)ws;  ws += sizeof(float) * (size_t)NEDGES;
    float* hproj  = (float*)ws;  ws += sizeof(float) * (size_t)NNODES * DIM;
    float* hagg   = (float*)ws;  ws += sizeof(float) * (size_t)NNODES * DIM;
    float* deg    = (float*)ws;  ws += sizeof(float) * (size_t)NNODES;
    float* concat = (float*)ws;  ws += sizeof(float) * (size_t)NNODES * FCIN;

    const int T = 256;
    const int gE  = (NEDGES + T - 1) / T;
    const int gN  = (NNODES + T - 1) / T;
    const int gND = (NNODES * DIM + T - 1) / T;
    const int gNF = (NNODES * FIN + T - 1) / T;
    const int gTiles = ((NNODES / 16) + 8 - 1) / 8;          // 8 waves/block
    const int gEw = (NEDGES + 8 - 1) / 8;                    // wave per edge
    const int gNw = (NNODES + 8 - 1) / 8;                    // wave per node

    // normalization, computed once (layer-invariant)
    hipMemsetAsync(deg, 0, sizeof(float) * (size_t)NNODES, stream);
    k_deg <<<gE, T, 0, stream>>>(col, ew, deg);
    k_dis <<<gN, T, 0, stream>>>(deg);
    k_norm<<<gE, T, 0, stream>>>(row, col, ew, deg, norm);

    // concat[:, :128] = x
    k_copyx<<<gNF, T, 0, stream>>>(x, concat);

    const float* Ws[3] = { W0, W1, W2 };
    const float* bs[3] = { b0, b1, b2 };
    for (int l = 0; l < 3; ++l) {
        const float* A = (l == 0) ? x : (concat + FIN + (l - 1) * DIM);
        const int lda  = (l == 0) ? FIN : FCIN;
        const int K    = (l == 0) ? FIN : DIM;
        k_gemm_wmma<<<gTiles, T, 0, stream>>>(A, lda, K, Ws[l], hproj);
        hipMemsetAsync(hagg, 0, sizeof(float) * (size_t)NNODES * DIM, stream);
        k_agg<<<gEw, T, 0, stream>>>(row, col, norm, hproj, hagg);
        k_bias_relu<<<gND, T, 0, stream>>>(hagg, bs[l], concat + FIN + l * DIM);
    }

    k_fc<<<gNw, T, 0, stream>>>(concat, fcW, fcb, out);
}